// UMixer_50311246905742
// MI455X (gfx1250) — compile-verified
//
#include <hip/hip_runtime.h>
#include <math.h>

typedef float v2f __attribute__((ext_vector_type(2)));
typedef float v8f __attribute__((ext_vector_type(8)));

#define BB 8
#define SEQ 336
#define PRED 96
#define CCH 321
#define DM 128
#define LTOT 432
#define NP 54      // patches for xemb
#define NPO 42     // patches for xold
#define NBC 2568   // B*C rows
#define ROWD 6912  // NP*DM

static __device__ __forceinline__ v8f wmma_f32(v2f a, v2f b, v8f c) {
  return __builtin_amdgcn_wmma_f32_16x16x4_f32(false, a, false, b, (short)0, c,
                                               false, false);
}

static __device__ __forceinline__ float gelu_exact(float x) {
  return 0.5f * x * (1.0f + erff(x * 0.70710678118654752440f));
}

// ---------------------------------------------------------------------------
// K1: xp[bc,l] = sum_s x[b,s,c] * pl_w[l,s] + pl_b[l]   (M=2568,N=432,K=336)
// ---------------------------------------------------------------------------
__global__ void k_proj(const float* __restrict__ x, const float* __restrict__ plw,
                       const float* __restrict__ plb, float* __restrict__ xp) {
  int lane = threadIdx.x;
  int lane16 = lane & 15;
  int khalf = (lane >> 4) * 2;
  int hi8 = (lane < 16) ? 0 : 8;
  int mt = blockIdx.x, nt = blockIdx.y;
  int mA = mt * 16 + lane16; if (mA > NBC - 1) mA = NBC - 1;
  int b = mA / CCH, c = mA - b * CCH;
  int abase = b * (SEQ * CCH) + c;
  int n = nt * 16 + lane16;  // l index, < 432 always
  v8f acc = {};
  for (int k0 = 0; k0 < SEQ; k0 += 4) {
    int ka = k0 + khalf;
    v2f a, bb2;
    a.x = x[abase + ka * CCH];
    a.y = x[abase + (ka + 1) * CCH];
    bb2.x = plw[n * SEQ + ka];
    bb2.y = plw[n * SEQ + ka + 1];
    acc = wmma_f32(a, bb2, acc);
  }
  float bias = plb[n];
  for (int r = 0; r < 8; ++r) {
    int mm = mt * 16 + r + hi8;
    if (mm < NBC) xp[mm * LTOT + n] = acc[r] + bias;
  }
}

// ---------------------------------------------------------------------------
// K2: patch embed (conv as GEMM, K=48) + positional encoding.
// mode 0: src = xp contiguous [n*432 + t], Lsrc=432, P=54
// mode 1: src = x_input transposed,        Lsrc=336, P=42
// ---------------------------------------------------------------------------
__global__ void k_pe(const float* __restrict__ src, const float* __restrict__ convw,
                     float* __restrict__ out, int P, int Lsrc, int mode) {
  int lane = threadIdx.x;
  int lane16 = lane & 15;
  int khalf = (lane >> 4) * 2;
  int hi8 = (lane < 16) ? 0 : 8;
  int n = blockIdx.x;
  int pt = blockIdx.y;
  int ot = blockIdx.z;
  int pA = pt * 16 + lane16; if (pA > P - 1) pA = P - 1;
  int o = ot * 16 + lane16;
  int base, stride;
  if (mode == 0) { base = n * LTOT; stride = 1; }
  else { int b = n / CCH, c = n - b * CCH; base = b * (SEQ * CCH) + c; stride = CCH; }
  v8f acc = {};
  for (int ks = 0; ks < 12; ++ks) {
    int ka = ks * 4 + khalf;
    v2f a, bb2;
    {
      int k = ka, kk = k >> 4, i = k & 15;
      int q = pA - 1 + kk; q = (q < 0) ? q + P : (q >= P ? q - P : q);
      int t = q * 8 + i; if (t > Lsrc - 1) t = Lsrc - 1;
      a.x = src[base + t * stride];
      bb2.x = convw[o * 48 + i * 3 + kk];
    }
    {
      int k = ka + 1, kk = k >> 4, i = k & 15;
      int q = pA - 1 + kk; q = (q < 0) ? q + P : (q >= P ? q - P : q);
      int t = q * 8 + i; if (t > Lsrc - 1) t = Lsrc - 1;
      a.y = src[base + t * stride];
      bb2.y = convw[o * 48 + i * 3 + kk];
    }
    acc = wmma_f32(a, bb2, acc);
  }
  int j2 = o & ~1;
  float div = expf((float)j2 * -0.0719557841560644f);  // -ln(10000)/128
  for (int r = 0; r < 8; ++r) {
    int pg = pt * 16 + r + hi8;
    if (pg < P) {
      float ang = (float)pg * div;
      float pe = (o & 1) ? cosf(ang) : sinf(ang);
      out[(n * P + pg) * DM + o] = acc[r] + pe;
    }
  }
}

// ---------------------------------------------------------------------------
// K3: temp-mix (pre-LN): dst = x + W2^T gelu(W1^T x + b1) + b2, per patch p.
// Block = 8 waves, each wave one 16-col tile of N=128; h staged in LDS.
// ---------------------------------------------------------------------------
__global__ void k_temp(const float* __restrict__ src, float* __restrict__ dst,
                       const float* __restrict__ w1, const float* __restrict__ b1,
                       const float* __restrict__ w2, const float* __restrict__ b2) {
  __shared__ float sh[16][DM];
  int tid = threadIdx.x;
  int wave = tid >> 5, lane = tid & 31;
  int lane16 = lane & 15;
  int khalf = (lane >> 4) * 2;
  int hi8 = (lane < 16) ? 0 : 8;
  int mt = blockIdx.x;
  int p = blockIdx.y;
  int og = wave * 16 + lane16;
  int mA = mt * 16 + lane16; if (mA > NBC - 1) mA = NBC - 1;
  const float* wp1 = w1 + p * (DM * DM);
  const float* wp2 = w2 + p * (DM * DM);
  v8f acc = {};
  for (int k0 = 0; k0 < DM; k0 += 4) {
    int ka = k0 + khalf;
    v2f a, bb2;
    a.x = src[mA * ROWD + p * DM + ka];
    a.y = src[mA * ROWD + p * DM + ka + 1];
    bb2.x = wp1[og * DM + ka];
    bb2.y = wp1[og * DM + ka + 1];
    acc = wmma_f32(a, bb2, acc);
  }
  float b1v = b1[p * DM + og];
  for (int r = 0; r < 8; ++r) sh[r + hi8][og] = gelu_exact(acc[r] + b1v);
  __syncthreads();
  v8f acc2 = {};
  for (int k0 = 0; k0 < DM; k0 += 4) {
    int ka = k0 + khalf;
    v2f a, bb2;
    a.x = sh[lane16][ka];
    a.y = sh[lane16][ka + 1];
    bb2.x = wp2[og * DM + ka];
    bb2.y = wp2[og * DM + ka + 1];
    acc2 = wmma_f32(a, bb2, acc2);
  }
  float b2v = b2[p * DM + og];
  for (int r = 0; r < 8; ++r) {
    int mm = mt * 16 + r + hi8;
    if (mm < NBC) {
      int idx = mm * ROWD + p * DM + og;
      dst[idx] = acc2[r] + b2v + src[idx];
    }
  }
}

// ---------------------------------------------------------------------------
// K4: chan-mix (pre-LN), per channel d: K=N=54 padded to 56/64.
// Block = 4 waves (N tiles of 16 up to 64); h staged zero-padded in LDS.
// ---------------------------------------------------------------------------
__global__ void k_chan(const float* __restrict__ src, float* __restrict__ dst,
                       const float* __restrict__ w1, const float* __restrict__ b1,
                       const float* __restrict__ w2, const float* __restrict__ b2) {
  __shared__ float sh[16][64];
  int tid = threadIdx.x;
  int wave = tid >> 5, lane = tid & 31;
  int lane16 = lane & 15;
  int khalf = (lane >> 4) * 2;
  int hi8 = (lane < 16) ? 0 : 8;
  int mt = blockIdx.x;
  int d = blockIdx.y;
  int qg = wave * 16 + lane16;
  int mA = mt * 16 + lane16; if (mA > NBC - 1) mA = NBC - 1;
  const float* wd1 = w1 + d * (NP * NP);
  const float* wd2 = w2 + d * (NP * NP);
  v8f acc = {};
  for (int k0 = 0; k0 < 56; k0 += 4) {
    int ka = k0 + khalf;
    int kc0 = (ka < NP) ? ka : NP - 1;
    int kc1 = (ka + 1 < NP) ? ka + 1 : NP - 1;
    v2f a, bb2;
    a.x = src[mA * ROWD + kc0 * DM + d];
    a.y = src[mA * ROWD + kc1 * DM + d];
    bb2.x = (ka < NP && qg < NP) ? wd1[qg * NP + ka] : 0.0f;
    bb2.y = (ka + 1 < NP && qg < NP) ? wd1[qg * NP + ka + 1] : 0.0f;
    acc = wmma_f32(a, bb2, acc);
  }
  float b1v = (qg < NP) ? b1[d * NP + qg] : 0.0f;
  for (int r = 0; r < 8; ++r)
    sh[r + hi8][qg] = (qg < NP) ? gelu_exact(acc[r] + b1v) : 0.0f;
  __syncthreads();
  v8f acc2 = {};
  for (int k0 = 0; k0 < 64; k0 += 4) {
    int ka = k0 + khalf;
    v2f a, bb2;
    a.x = sh[lane16][ka];
    a.y = sh[lane16][ka + 1];
    bb2.x = (ka < NP && qg < NP) ? wd2[qg * NP + ka] : 0.0f;
    bb2.y = (ka + 1 < NP && qg < NP) ? wd2[qg * NP + ka + 1] : 0.0f;
    acc2 = wmma_f32(a, bb2, acc2);
  }
  if (qg < NP) {
    float b2v = b2[d * NP + qg];
    for (int r = 0; r < 8; ++r) {
      int mm = mt * 16 + r + hi8;
      if (mm < NBC) {
        int idx = mm * ROWD + qg * DM + d;
        dst[idx] = acc2[r] + b2v + src[idx];
      }
    }
  }
}

// ---------------------------------------------------------------------------
// K5: in-place LayerNorm over 128-element rows (wave32 shuffle reduction).
// ---------------------------------------------------------------------------
__global__ void k_ln(float* __restrict__ buf, const float* __restrict__ g,
                     const float* __restrict__ bsh, int nrows) {
  int wave = threadIdx.x >> 5, lane = threadIdx.x & 31;
  int row = blockIdx.x * 8 + wave;
  if (row >= nrows) return;
  int base = row * DM;
  float v[4];
  float s = 0.0f, q = 0.0f;
  for (int j = 0; j < 4; ++j) {
    v[j] = buf[base + lane + 32 * j];
    s += v[j]; q += v[j] * v[j];
  }
  for (int off = 16; off > 0; off >>= 1) {
    s += __shfl_xor(s, off, 32);
    q += __shfl_xor(q, off, 32);
  }
  float mean = s * (1.0f / 128.0f);
  float var = q * (1.0f / 128.0f) - mean * mean;
  float inv = rsqrtf(var + 1e-5f);
  for (int j = 0; j < 4; ++j) {
    int c = lane + 32 * j;
    buf[base + c] = (v[j] - mean) * inv * g[c] + bsh[c];
  }
}

// ---------------------------------------------------------------------------
// K6: s_corr. One block per n: LN(xold) and LN(comb*cw+cb) into LDS, then
// time-domain circular autocorrelation (== ortho rfft/irfft path), alpha out.
// ---------------------------------------------------------------------------
__global__ void k_corr(const float* __restrict__ xold, const float* __restrict__ comb,
                       const float* __restrict__ lng, const float* __restrict__ lnb,
                       const float* __restrict__ cwp, const float* __restrict__ cbp,
                       float* __restrict__ alpha) {
  __shared__ float sa[NPO * DM];
  __shared__ float sb[NPO * DM];
  int tid = threadIdx.x;
  int wave = tid >> 5, lane = tid & 31;
  int n = blockIdx.x;
  float cw = cwp[0], cb = cbp[0];
  for (int r = wave; r < NPO; r += 8) {
    float xv[4], pv[4];
    float sx = 0, sxx = 0, sp = 0, spp = 0;
    for (int j = 0; j < 4; ++j) {
      int c = lane + 32 * j;
      xv[j] = xold[(n * NPO + r) * DM + c];
      pv[j] = comb[n * ROWD + r * DM + c] * cw + cb;
      sx += xv[j]; sxx += xv[j] * xv[j];
      sp += pv[j]; spp += pv[j] * pv[j];
    }
    for (int off = 16; off > 0; off >>= 1) {
      sx += __shfl_xor(sx, off, 32); sxx += __shfl_xor(sxx, off, 32);
      sp += __shfl_xor(sp, off, 32); spp += __shfl_xor(spp, off, 32);
    }
    float mx = sx * (1.0f / 128.0f), vx = sxx * (1.0f / 128.0f) - mx * mx;
    float ix = rsqrtf(vx + 1e-5f);
    float mp = sp * (1.0f / 128.0f), vp = spp * (1.0f / 128.0f) - mp * mp;
    float ip = rsqrtf(vp + 1e-5f);
    for (int j = 0; j < 4; ++j) {
      int c = lane + 32 * j;
      float gg = lng[c], bb = lnb[c];
      sa[r * DM + c] = (xv[j] - mx) * ix * gg + bb;
      sb[r * DM + c] = (pv[j] - mp) * ip * gg + bb;
    }
  }
  __syncthreads();
  if (tid < DM) {
    int d = tid;
    float num = 0.0f, den = 0.0f;
    for (int k = 0; k < NPO; ++k) {
      float rx = 0.0f, rp = 0.0f;
      for (int p = 0; p < NPO; ++p) {
        int p2 = p + k; if (p2 >= NPO) p2 -= NPO;
        rx += sa[p * DM + d] * sa[p2 * DM + d];
        rp += sb[p * DM + d] * sb[p2 * DM + d];
      }
      rx = fmaxf(rx, 0.0f);
      rp = fmaxf(rp, 0.0f);
      num += rx * rp;
      den += rp * rp;
    }
    float invN = 1.0f / (float)NPO;
    alpha[n * DM + d] = sqrtf((num * invN) / (den * invN + 0.001f));
  }
}

// ---------------------------------------------------------------------------
// K7: head GEMM. A[n,k=d*54+p] = alpha[n,d]*(comb[n,p,d]*cw+cb); B = head_w.
// M=2568, N=96, K=6912. Output: out[b, t, c].
// ---------------------------------------------------------------------------
__global__ void k_head(const float* __restrict__ comb, const float* __restrict__ alpha,
                       const float* __restrict__ cwp, const float* __restrict__ cbp,
                       const float* __restrict__ hw, const float* __restrict__ hb,
                       float* __restrict__ out) {
  int lane = threadIdx.x;
  int lane16 = lane & 15;
  int khalf = (lane >> 4) * 2;
  int hi8 = (lane < 16) ? 0 : 8;
  int mt = blockIdx.x, nt = blockIdx.y;
  int mA = mt * 16 + lane16; if (mA > NBC - 1) mA = NBC - 1;
  int t = nt * 16 + lane16;  // < 96 always
  float cw = cwp[0], cb = cbp[0];
  v8f acc = {};
  for (int k0 = 0; k0 < ROWD; k0 += 4) {
    int ka = k0 + khalf;
    v2f a, bb2;
    {
      int d = ka / NP, p = ka - d * NP;
      a.x = alpha[mA * DM + d] * (comb[mA * ROWD + p * DM + d] * cw + cb);
    }
    {
      int k = ka + 1;
      int d = k / NP, p = k - d * NP;
      a.y = alpha[mA * DM + d] * (comb[mA * ROWD + p * DM + d] * cw + cb);
    }
    bb2.x = hw[t * ROWD + ka];
    bb2.y = hw[t * ROWD + ka + 1];
    acc = wmma_f32(a, bb2, acc);
  }
  float bias = hb[t];
  for (int r = 0; r < 8; ++r) {
    int mm = mt * 16 + r + hi8;
    if (mm < NBC) {
      int b = mm / CCH, c = mm - b * CCH;
      out[b * (PRED * CCH) + t * CCH + c] = acc[r] + bias;
    }
  }
}

// ---------------------------------------------------------------------------
extern "C" void kernel_launch(void* const* d_in, const int* in_sizes, int n_in,
                              void* d_out, int out_size, void* d_ws, size_t ws_size,
                              hipStream_t stream) {
  (void)in_sizes; (void)n_in; (void)out_size; (void)ws_size;
  const float* x      = (const float*)d_in[0];
  const float* pl_w   = (const float*)d_in[1];
  const float* pl_b   = (const float*)d_in[2];
  const float* conv_w = (const float*)d_in[3];
  const float* tmd_w1 = (const float*)d_in[4];
  const float* tmd_b1 = (const float*)d_in[5];
  const float* tmd_w2 = (const float*)d_in[6];
  const float* tmd_b2 = (const float*)d_in[7];
  const float* tmd_lg = (const float*)d_in[8];
  const float* tmd_lb = (const float*)d_in[9];
  const float* tmu_w1 = (const float*)d_in[10];
  const float* tmu_b1 = (const float*)d_in[11];
  const float* tmu_w2 = (const float*)d_in[12];
  const float* tmu_b2 = (const float*)d_in[13];
  const float* tmu_lg = (const float*)d_in[14];
  const float* tmu_lb = (const float*)d_in[15];
  const float* cmd_w1 = (const float*)d_in[16];
  const float* cmd_b1 = (const float*)d_in[17];
  const float* cmd_w2 = (const float*)d_in[18];
  const float* cmd_b2 = (const float*)d_in[19];
  const float* cmd_lg = (const float*)d_in[20];
  const float* cmd_lb = (const float*)d_in[21];
  const float* cmu_w1 = (const float*)d_in[22];
  const float* cmu_b1 = (const float*)d_in[23];
  const float* cmu_w2 = (const float*)d_in[24];
  const float* cmu_b2 = (const float*)d_in[25];
  const float* cmu_lg = (const float*)d_in[26];
  const float* cmu_lb = (const float*)d_in[27];
  const float* ln_g   = (const float*)d_in[28];
  const float* ln_b   = (const float*)d_in[29];
  const float* comb_w = (const float*)d_in[30];
  const float* comb_b = (const float*)d_in[31];
  const float* head_w = (const float*)d_in[32];
  const float* head_b = (const float*)d_in[33];
  float* out = (float*)d_out;

  float* ws = (float*)d_ws;
  const size_t S_xp  = (size_t)NBC * LTOT;        // 1,109,376
  const size_t S_emb = (size_t)NBC * NP * DM;     // 17,750,016
  const size_t S_old = (size_t)NBC * NPO * DM;    // 13,805,568
  float* xp    = ws;
  float* xemb  = xp + S_xp;
  float* ping  = xemb + S_emb;
  float* pong  = ping + S_emb;
  float* xold  = pong + S_emb;
  float* alpha = xold + S_old;

  // layer strides
  const size_t TW = (size_t)NP * DM * DM;  // temp w per layer
  const size_t TBS = (size_t)NP * DM;      // temp b per layer
  const size_t CW = (size_t)DM * NP * NP;  // chan w per layer
  const size_t CBS = (size_t)DM * NP;      // chan b per layer

  const int MT = (NBC + 15) / 16;  // 161
  const int NROWS = NBC * NP;      // 138,672
  dim3 lnGrid((NROWS + 7) / 8);

  // 1. projection GEMM
  k_proj<<<dim3(MT, LTOT / 16), 32, 0, stream>>>(x, pl_w, pl_b, xp);
  // 2. patch embeds
  k_pe<<<dim3(NBC, 4, 8), 32, 0, stream>>>(xp, conv_w, xemb, NP, LTOT, 0);
  k_pe<<<dim3(NBC, 3, 8), 32, 0, stream>>>(x, conv_w, xold, NPO, SEQ, 1);

  // 3. mixer chain (only li = E-1 survives): layer1 down, layer1..0 up
  // TM(tmd[1])
  k_temp<<<dim3(MT, NP), 256, 0, stream>>>(xemb, ping, tmd_w1 + TW, tmd_b1 + TBS,
                                           tmd_w2 + TW, tmd_b2 + TBS);
  k_ln<<<lnGrid, 256, 0, stream>>>(ping, tmd_lg + DM, tmd_lb + DM, NROWS);
  // CM(cmd[1])
  k_chan<<<dim3(MT, DM), 128, 0, stream>>>(ping, pong, cmd_w1 + CW, cmd_b1 + CBS,
                                           cmd_w2 + CW, cmd_b2 + CBS);
  k_ln<<<lnGrid, 256, 0, stream>>>(pong, cmd_lg + DM, cmd_lb + DM, NROWS);
  // TM(tmu[1])
  k_temp<<<dim3(MT, NP), 256, 0, stream>>>(pong, ping, tmu_w1 + TW, tmu_b1 + TBS,
                                           tmu_w2 + TW, tmu_b2 + TBS);
  k_ln<<<lnGrid, 256, 0, stream>>>(ping, tmu_lg + DM, tmu_lb + DM, NROWS);
  // CM(cmu[1])
  k_chan<<<dim3(MT, DM), 128, 0, stream>>>(ping, pong, cmu_w1 + CW, cmu_b1 + CBS,
                                           cmu_w2 + CW, cmu_b2 + CBS);
  k_ln<<<lnGrid, 256, 0, stream>>>(pong, cmu_lg + DM, cmu_lb + DM, NROWS);
  // TM(tmu[0])
  k_temp<<<dim3(MT, NP), 256, 0, stream>>>(pong, ping, tmu_w1, tmu_b1, tmu_w2, tmu_b2);
  k_ln<<<lnGrid, 256, 0, stream>>>(ping, tmu_lg, tmu_lb, NROWS);
  // CM(cmu[0])
  k_chan<<<dim3(MT, DM), 128, 0, stream>>>(ping, pong, cmu_w1, cmu_b1, cmu_w2, cmu_b2);
  k_ln<<<lnGrid, 256, 0, stream>>>(pong, cmu_lg, cmu_lb, NROWS);

  // 4. s_corr -> alpha
  k_corr<<<dim3(NBC), 256, 0, stream>>>(xold, pong, ln_g, ln_b, comb_w, comb_b, alpha);

  // 5. head GEMM -> output
  k_head<<<dim3(MT, PRED / 16), 32, 0, stream>>>(pong, alpha, comb_w, comb_b,
                                                 head_w, head_b, out);
}